// CDifferentialMaxtree_86887188398633
// MI455X (gfx1250) — compile-verified
//
#include <hip/hip_runtime.h>
#include <stdint.h>

typedef __attribute__((ext_vector_type(2))) float v2f;
typedef __attribute__((ext_vector_type(8))) float v8f;

#define P_NODES   (1 << 20)        // H*W = 1024*1024
#define PP_STRIDE (P_NODES + 4)    // sentinel slot at index P, padded for 16B alignment
#define N_NETS    2                // N
#define N_TREES   4                // B*N
#define TILE      256
#define DOUBLING_ITERS 20          // ceil(log2(P))

// ---------------- CDNA5 async global->LDS copy (ASYNCcnt path) ----------------
__device__ __forceinline__ void async_copy_b128_to_lds(uint32_t lds_off, const void* gaddr) {
  asm volatile("global_load_async_to_lds_b128 %0, %1, off"
               :: "v"(lds_off), "v"(gaddr)
               : "memory");
}
__device__ __forceinline__ void wait_asynccnt0() {
  asm volatile("s_wait_asynccnt 0" ::: "memory");
}

// ---------------------------------------------------------------------------
// Kernel A: feature engineering + WMMA GEMV + sigmoid + per-node contribution.
// Writes contrib -> v0, modified parents (root -> sentinel P) -> p0,
// and initializes sentinel slots in all four ping-pong buffers.
// ---------------------------------------------------------------------------
__global__ __launch_bounds__(TILE) void contrib_kernel(
    const float* __restrict__ x, const float* __restrict__ weight,
    const float* __restrict__ bias, const float* __restrict__ attrs,
    const int* __restrict__ parents,
    float* __restrict__ v0, float* __restrict__ v1,
    int* __restrict__ p0, int* __restrict__ p1)
{
  __shared__ float sAttr[TILE * 15];  // raw attrs tile (15360 B), async-staged
  __shared__ float sF[TILE * 20];     // 17 engineered feats + 3 zero pad (stride 20: bank-clean)
  __shared__ float sW[20];            // padded weight vector (keeps B-fragment out of VGPR extracts)
  __shared__ float sLin[TILE];        // GEMV results

  const int tid = threadIdx.x;
  const int tile = blockIdx.x;
  const int tree = tile >> 12;                    // P/TILE = 4096 tiles per tree
  const int net = tree & (N_NETS - 1);            // tree index = b*N + n
  const int pBase = (tile & 4095) * TILE;         // node index within tree
  const long long gBase = (long long)tree * P_NODES + pBase;

  // ---- async stage: 256 nodes * 60 B = 15360 B = 960 x b128, contiguous ----
  {
    const char* src = (const char*)attrs + gBase * 60;
    const uint32_t dst = (uint32_t)(size_t)&sAttr[0];
#pragma unroll
    for (int r = 0; r < 3; ++r) {
      const int e = tid + r * TILE;
      async_copy_b128_to_lds(dst + e * 16, src + e * 16);
    }
    if (tid < 192) {
      const int e = tid + 3 * TILE;
      async_copy_b128_to_lds(dst + e * 16, src + e * 16);
    }
    wait_asynccnt0();
  }

  // ---- padded weight vector -> LDS (uniform per block) ----
  if (tid < 20) {
    sW[tid] = (tid < 17) ? weight[net * 17 + tid] : 0.0f;
  }
  __syncthreads();

  // ---- per-node feature engineering (matches _log_scaling ordering) ----
  {
    const float* a = &sAttr[tid * 15];
    float* F = &sF[tid * 20];
#pragma unroll
    for (int j = 0; j < 5; ++j) F[j] = a[j];                 // xmin..area
#pragma unroll
    for (int j = 0; j < 9; ++j)                               // log|cols 6..14|
      F[5 + j] = __logf(fabsf(a[6 + j]) + 1e-10f);
    F[14] = sqrtf(a[7]) / (sqrtf(a[6]) + 1e-10f);             // lshape
    float sn, cs;
    __sincosf(a[5], &sn, &cs);
    F[15] = cs;                                               // cos(angle)
    F[16] = sn;                                               // sin(angle)
    F[17] = 0.0f; F[18] = 0.0f; F[19] = 0.0f;                 // K pad to 20
  }
  __syncthreads();

  // ---- 17->1 GEMV on the matrix pipe: 5 x V_WMMA_F32_16X16X4_F32 per batch ----
  const int lane = tid & 31;
  const int waveBase = tid & ~31;
  const int row = lane & 15;
  const int khi = (lane & 16) ? 2 : 0; // lanes 0-15 carry K+0/K+1, lanes 16-31 K+2/K+3

  // Runtime-offset LDS base pointers -> fragment loads are immediate-offset ds_loads.
  const float* Ar0 = &sF[(waveBase + row) * 20 + khi];
  const float* Ar1 = &sF[(waveBase + 16 + row) * 20 + khi];
  const float* Wr  = &sW[khi];   // B[k][j] = w[k] for all 16 columns

  v8f acc0 = {0.f,0.f,0.f,0.f,0.f,0.f,0.f,0.f};
  v8f acc1 = {0.f,0.f,0.f,0.f,0.f,0.f,0.f,0.f};
#pragma unroll
  for (int c = 0; c < 5; ++c) {
    v2f bf; bf.x = Wr[4 * c];  bf.y = Wr[4 * c + 1];
    v2f a0; a0.x = Ar0[4 * c]; a0.y = Ar0[4 * c + 1];
    v2f a1; a1.x = Ar1[4 * c]; a1.y = Ar1[4 * c + 1];
    acc0 = __builtin_amdgcn_wmma_f32_16x16x4_f32(false, a0, false, bf, (short)0, acc0, false, false);
    acc1 = __builtin_amdgcn_wmma_f32_16x16x4_f32(false, a1, false, bf, (short)0, acc1, false, false);
  }
  // All D columns are identical; scatter column N=0 back to thread order.
  if (row == 0) {                       // lanes 0 (rows 0-7) and 16 (rows 8-15)
    const int half = (lane & 16) ? 8 : 0;
#pragma unroll
    for (int r = 0; r < 8; ++r) {
      sLin[waveBase + half + r]      = acc0[r];
      sLin[waveBase + 16 + half + r] = acc1[r];
    }
  }
  __syncthreads();

  // ---- sigmoid, gray-level increment, contribution ----
  const float lin = sLin[tid] + bias[net];
  const float score = 1.0f / (1.0f + __expf(-lin));

  const long long g = gBase + tid;
  const int pl = pBase + tid;
  const float xv = x[g];
  const int par = parents[g];
  const bool root = (par == pl);
  const float pv = x[(long long)tree * P_NODES + par];   // L2-resident gather
  const float contrib = (root ? xv : (xv - pv)) * score;

  const size_t base = (size_t)tree * PP_STRIDE;
  v0[base + pl] = contrib;
  p0[base + pl] = root ? P_NODES : par;
  if (pl == 0) {  // sentinel: v=0, p=P in both ping-pong buffers
    v0[base + P_NODES] = 0.0f; v1[base + P_NODES] = 0.0f;
    p0[base + P_NODES] = P_NODES; p1[base + P_NODES] = P_NODES;
  }
}

// ---------------------------------------------------------------------------
// Kernel B: one pointer-doubling pass: v' = v + v[p]; p' = p[p].
// 4 nodes/thread with b128 streams; gathers hit L2 (34 MB working set).
// Last iteration writes d_out directly.
// ---------------------------------------------------------------------------
__global__ __launch_bounds__(TILE) void pointer_double_kernel(
    const float* __restrict__ vs, const int* __restrict__ ps,
    float* __restrict__ vd, int* __restrict__ pdst,
    float* __restrict__ out, const int last)
{
  const long long node0 = ((long long)blockIdx.x * TILE + threadIdx.x) * 4;
  const int tree = (int)(node0 >> 20);
  const int p = (int)(node0 & (P_NODES - 1));
  const size_t base = (size_t)tree * PP_STRIDE;

  const float4 v4 = *(const float4*)(vs + base + p);
  const int4  q4 = *(const int4*)(ps + base + p);

  float4 r;
  r.x = v4.x + vs[base + q4.x];
  r.y = v4.y + vs[base + q4.y];
  r.z = v4.z + vs[base + q4.z];
  r.w = v4.w + vs[base + q4.w];

  if (last) {
    *(float4*)(out + node0) = r;
  } else {
    int4 np;
    np.x = ps[base + q4.x];
    np.y = ps[base + q4.y];
    np.z = ps[base + q4.z];
    np.w = ps[base + q4.w];
    *(float4*)(vd + base + p) = r;
    *(int4*)(pdst + base + p) = np;
  }
}

extern "C" void kernel_launch(void* const* d_in, const int* in_sizes, int n_in,
                              void* d_out, int out_size, void* d_ws, size_t ws_size,
                              hipStream_t stream) {
  const float* x       = (const float*)d_in[0];
  const float* weight  = (const float*)d_in[1];
  const float* bias    = (const float*)d_in[2];
  const float* attrs   = (const float*)d_in[3];
  const int*   parents = (const int*)d_in[4];
  float* out = (float*)d_out;

  // workspace: v ping/pong + p ping/pong, each N_TREES*PP_STRIDE elems (~67 MB total)
  const size_t arr = (size_t)N_TREES * PP_STRIDE;
  float* v0 = (float*)d_ws;
  float* v1 = v0 + arr;
  int* p0 = (int*)(v1 + arr);
  int* p1 = p0 + arr;

  const int tiles = N_TREES * (P_NODES / TILE);            // 16384
  hipLaunchKernelGGL(contrib_kernel, dim3(tiles), dim3(TILE), 0, stream,
                     x, weight, bias, attrs, parents, v0, v1, p0, p1);

  const int blocks = (N_TREES * P_NODES) / (4 * TILE);     // 4096
  for (int i = 0; i < DOUBLING_ITERS; ++i) {
    const bool even = ((i & 1) == 0);
    const float* vs = even ? v0 : v1;
    const int*   ps = even ? p0 : p1;
    float* vd = even ? v1 : v0;
    int*   pn = even ? p1 : p0;
    const int last = (i == DOUBLING_ITERS - 1) ? 1 : 0;
    hipLaunchKernelGGL(pointer_double_kernel, dim3(blocks), dim3(TILE), 0, stream,
                       vs, ps, vd, pn, out, last);
  }
}